// MaskedMedianPool1d_11046655885419
// MI455X (gfx1250) — compile-verified
//
#include <hip/hip_runtime.h>
#include <hip/hip_bf16.h>
#include <math.h>

// MaskedMedianPool1d: B=16, C=128, W=8192, K=7, PAD=3, STRIDE=1
// out[b,c,o] = lower-median of valid entries in x[b,c,o-3 .. o+3],
// valid = in-range && mask; NaN if no valid entries. Reflect padding in the
// reference is dead logic (padded positions have mask==0 -> +inf -> never picked).

#define KSZ    7
#define PADW   3
#define BDIM   256
#define SEG    1024                 // outputs per block
#define WROW   8192
#define NSEG   (WROW / SEG)         // 8
#define TILEA  1032                 // aligned tile: [seg*SEG - 4, seg*SEG + 1028)
#define NCHUNK (TILEA / 4)          // 258 float4/int4 chunks

__device__ __forceinline__ void ce(float &a, float &b) {
    float lo = fminf(a, b);
    float hi = fmaxf(a, b);
    a = lo; b = hi;
}

__global__ __launch_bounds__(BDIM) void masked_median_pool1d_kernel(
    const float* __restrict__ x,
    const int*   __restrict__ mask,
    float*       __restrict__ out)
{
    __shared__ float xb[TILEA];     // raw x tile   (async DMA target)
    __shared__ int   mb[TILEA];     // raw mask tile (async DMA target)

    const int tid = threadIdx.x;
    const int row = (int)blockIdx.x >> 3;        // b*C + c  (2048 rows)
    const int seg = (int)blockIdx.x & (NSEG - 1);
    const long long rbase = (long long)row * WROW;
    const int a0 = seg * SEG - 4;                // global col of xb[0], 4-elem aligned

    const unsigned long long xbase = (unsigned long long)(uintptr_t)x;
    const unsigned long long mbase = (unsigned long long)(uintptr_t)mask;

    // ---- Load phase: async global->LDS b128 DMA, one HBM touch per byte ----
    // a0 % 4 == 0 and WROW % 4 == 0 => out-of-range happens only as whole chunks.
    for (int c = tid; c < NCHUNK; c += BDIM) {
        const int lo = a0 + 4 * c;
        // LDS flat pointers: low 32 bits == LDS byte offset (ISA aperture rule)
        const unsigned int xoff = (unsigned int)(uintptr_t)&xb[4 * c];
        const unsigned int moff = (unsigned int)(uintptr_t)&mb[4 * c];
        if (lo >= 0 && lo + 3 < WROW) {
            const int gbyte = (int)((rbase + lo) * 4); // < 2^27, fits i32
            asm volatile("global_load_async_to_lds_b128 %0, %1, %2"
                         :: "v"(xoff), "v"(gbyte), "s"(xbase) : "memory");
            asm volatile("global_load_async_to_lds_b128 %0, %1, %2"
                         :: "v"(moff), "v"(gbyte), "s"(mbase) : "memory");
        } else {
            // fully out-of-range chunk: mask=0 -> window entry becomes +inf
            *reinterpret_cast<int4*>(&mb[4 * c])   = make_int4(0, 0, 0, 0);
            *reinterpret_cast<float4*>(&xb[4 * c]) = make_float4(0.f, 0.f, 0.f, 0.f);
        }
    }
    // wait for this wave's async DMA into LDS, then make it workgroup-visible
    asm volatile("s_wait_asynccnt 0x0" ::: "memory");
    __syncthreads();

    // ---- Compute phase: 4 consecutive outputs per thread ----
    // output o (local) uses global cols seg*SEG + o - 3 + k = a0 + (o+1) + k
    const int o0 = tid * 4;                      // local outputs o0..o0+3
    // aligned LDS reads of [o0 .. o0+11]  (3x ds_load_b128 each)
    float xr[12];
    int   mr[12];
    *reinterpret_cast<float4*>(&xr[0]) = *reinterpret_cast<const float4*>(&xb[o0]);
    *reinterpret_cast<float4*>(&xr[4]) = *reinterpret_cast<const float4*>(&xb[o0 + 4]);
    *reinterpret_cast<float4*>(&xr[8]) = *reinterpret_cast<const float4*>(&xb[o0 + 8]);
    *reinterpret_cast<int4*>(&mr[0])   = *reinterpret_cast<const int4*>(&mb[o0]);
    *reinterpret_cast<int4*>(&mr[4])   = *reinterpret_cast<const int4*>(&mb[o0 + 4]);
    *reinterpret_cast<int4*>(&mr[8])   = *reinterpret_cast<const int4*>(&mb[o0 + 8]);

    const float INF = __builtin_inff();
    float w[12];
    #pragma unroll
    for (int j = 1; j <= 10; ++j)                // only indices 1..10 are used
        w[j] = (mr[j] != 0) ? xr[j] : INF;       // v_cndmask

    float4 res;
    float* rp = &res.x;

    #pragma unroll
    for (int k = 0; k < 4; ++k) {
        float s0 = w[k + 1], s1 = w[k + 2], s2 = w[k + 3], s3 = w[k + 4];
        float s4 = w[k + 5], s5 = w[k + 6], s6 = w[k + 7];

        // valid count (masked entries are exactly +inf; x is finite)
        int cnt = (s0 < INF) + (s1 < INF) + (s2 < INF) + (s3 < INF)
                + (s4 < INF) + (s5 < INF) + (s6 < INF);

        // odd-even transposition sort, 7 rounds (guaranteed correct, 21 CE)
        ce(s0,s1); ce(s2,s3); ce(s4,s5);
        ce(s1,s2); ce(s3,s4); ce(s5,s6);
        ce(s0,s1); ce(s2,s3); ce(s4,s5);
        ce(s1,s2); ce(s3,s4); ce(s5,s6);
        ce(s0,s1); ce(s2,s3); ce(s4,s5);
        ce(s1,s2); ce(s3,s4); ce(s5,s6);
        ce(s0,s1); ce(s2,s3); ce(s4,s5);
        // s0..s3 now hold the 4 smallest in order; idx=(cnt-1)>>1 <= 3

        const int idx = (cnt - 1) >> 1;
        float r = s0;
        r = (idx >= 1) ? s1 : r;
        r = (idx >= 2) ? s2 : r;
        r = (idx >= 3) ? s3 : r;
        if (cnt == 0) r = __builtin_nanf("");
        rp[k] = r;
    }

    // single 16B coalesced store (global_store_b128)
    *reinterpret_cast<float4*>(&out[rbase + seg * SEG + o0]) = res;
}

extern "C" void kernel_launch(void* const* d_in, const int* in_sizes, int n_in,
                              void* d_out, int out_size, void* d_ws, size_t ws_size,
                              hipStream_t stream) {
    const float* x    = (const float*)d_in[0];
    const int*   mask = (const int*)d_in[1];
    float*       out  = (float*)d_out;

    const int rows = 16 * 128;            // B*C
    dim3 grid(rows * NSEG);               // 16384 blocks
    dim3 block(BDIM);                     // 8 wave32 per block
    masked_median_pool1d_kernel<<<grid, block, 0, stream>>>(x, mask, out);
}